// AdaptiveStateManager_82136954569341
// MI455X (gfx1250) — compile-verified
//
#include <hip/hip_runtime.h>
#include <cstddef>
#include <cstdint>

// Problem constants (match reference)
#define Bsz 64
#define Ssz 512
#define Hsz 1024
#define H4  4096
#define BH  (Bsz * Hsz)

// LDS staging geometry for the scan kernels
#define KC      128                 // K-chunk per stage
#define BSTRIDE 132                 // padded floats per LDS row (132 % 64 == 4 -> conflict-free v2f reads)
#define BUFSZ   (64 * BSTRIDE)      // 4 gates x 16 rows
#define NCHUNK  (Hsz / KC)

typedef __attribute__((ext_vector_type(2))) float v2f;
typedef __attribute__((ext_vector_type(8))) float v8f;
typedef __attribute__((ext_vector_type(4))) int   v4i;

__device__ __forceinline__ float sigf(float x) { return 1.0f / (1.0f + __expf(-x)); }

// ---------------------------------------------------------------------------
// FP32 WMMA helper: D = A(16x4) * B(4x16) + C   (V_WMMA_F32_16X16X4_F32)
// A layout: lanes 0-15 hold rows M=0..15 with {K0,K1}; lanes 16-31 hold {K2,K3}
// B layout (mirror over N): lanes 0-15 hold cols N=0..15 with {K0,K1}; 16-31 {K2,K3}
// C/D layout: VGPR j: lanes 0-15 -> (M=j, N=lane); lanes 16-31 -> (M=8+j, N=lane-16)
// ---------------------------------------------------------------------------
__device__ __forceinline__ v8f wmma_f32(v2f a, v2f b, v8f c) {
    return __builtin_amdgcn_wmma_f32_16x16x4_f32(false, a, false, b, (short)0, c,
                                                 false, false);
}

// ---------------------------------------------------------------------------
// Async global -> LDS staging (CDNA5, ASYNCcnt-tracked). The builtin takes
// (v4i AS1* src, v4i AS3* dst, imm offset, imm cpol). LDS address is derived
// via the low-32-bit LDS-aperture truncation of the generic pointer.
// Guarded: falls back to plain ld/st if the builtin is absent.
// ---------------------------------------------------------------------------
#if __has_builtin(__builtin_amdgcn_global_load_async_to_lds_b128)
#define HAVE_ASYNC_LDS 1
typedef __attribute__((address_space(1))) v4i* gbl_v4i_t;
typedef __attribute__((address_space(3))) v4i* lds_v4i_t;
#else
#define HAVE_ASYNC_LDS 0
#endif

// Copy 64 contiguous floats (16B-aligned) from global to LDS for this thread.
__device__ __forceinline__ void stage_seg64(const float* __restrict__ gsrc,
                                            float* __restrict__ ldst) {
#if HAVE_ASYNC_LDS
    gbl_v4i_t gp = (gbl_v4i_t)(uintptr_t)gsrc;
    lds_v4i_t lp = (lds_v4i_t)(uint32_t)(uintptr_t)ldst;
#pragma unroll
    for (int i = 0; i < 16; ++i)
        __builtin_amdgcn_global_load_async_to_lds_b128(gp + i, lp + i, 0, 0);
#else
#pragma unroll
    for (int i = 0; i < 16; ++i)
        *(float4*)(ldst + 4 * i) = *(const float4*)(gsrc + 4 * i);
#endif
}

__device__ __forceinline__ void wait_async_all() {
#if HAVE_ASYNC_LDS
#if __has_builtin(__builtin_amdgcn_s_wait_asynccnt)
    __builtin_amdgcn_s_wait_asynccnt(0);
#else
    asm volatile("s_wait_asynccnt 0" ::: "memory");
#endif
#endif
}

// ---------------------------------------------------------------------------
// Accumulate 4 gate tiles: acc[g] += A(16 rows of this wave) @ W[g-block]^T
// W rows for this block's nt are staged into LDS (double-buffered, async),
// shared by all 4 waves (one wave per 16-row m-tile).
//   W : [4H, H] row-major, rows used: g*H + n0 + r, r in 0..15
//   A : [64, H] row-major (rows m = wave*16 .. +15)
// ---------------------------------------------------------------------------
__device__ __forceinline__ void accum_gates_lds(v8f* acc, const float* __restrict__ W,
                                                const float* __restrict__ A,
                                                float* sB, int n0) {
    const int tid  = threadIdx.x;
    const int lane = tid & 31;
    const int half = lane >> 4;
    const int l16  = lane & 15;
    const int wave = tid >> 5;  // m-tile of this wave

    // staging map: 64 LDS rows (g,r), 2 threads per row, 64 floats each
    const int srow = tid >> 1;          // 0..63 : g = srow>>4, r = srow&15
    const int sseg = (tid & 1) * 64;    // column segment within the chunk
    const int g_s  = srow >> 4;
    const int r_s  = srow & 15;
    const float* gsrc = W + (size_t)(g_s * Hsz + n0 + r_s) * Hsz + sseg;
    const int    loff = srow * BSTRIDE + sseg;

    const float* arow = A + (size_t)(wave * 16 + l16) * Hsz;

    // prologue: stage chunk 0 into buffer 0
    stage_seg64(gsrc, sB + loff);

    for (int kc = 0; kc < NCHUNK; ++kc) {
        wait_async_all();
        __syncthreads();  // chunk kc staged for all waves; prev readers done

        if (kc + 1 < NCHUNK)
            stage_seg64(gsrc + (kc + 1) * KC, sB + ((kc + 1) & 1) * BUFSZ + loff);

        const float* buf = sB + (kc & 1) * BUFSZ;
        const float* ab  = arow + kc * KC;
        for (int kk = 0; kk < KC; kk += 4) {
            v2f a = *(const v2f*)(ab + kk + 2 * half);
#pragma unroll
            for (int g = 0; g < 4; ++g) {
                v2f b = *(const v2f*)(buf + (g * 16 + l16) * BSTRIDE + kk + 2 * half);
                acc[g] = wmma_f32(a, b, acc[g]);
            }
        }
    }
}

// ===========================================================================
// Phase 1: G0 = X @ W_ih0^T + (b_ih0 + b_hh0)
//   X : [B*S, H], W_ih0 : [4H, H], G0 : [B*S, 4H]
// Each wave computes a 16x64 output tile (4 WMMA accumulators sharing A).
// ===========================================================================
__global__ __launch_bounds__(128) void g0_gemm_kernel(
    const float* __restrict__ X, const float* __restrict__ Wih0,
    const float* __restrict__ bih0, const float* __restrict__ bhh0,
    float* __restrict__ G0) {
    const int lane = threadIdx.x & 31;
    const int half = lane >> 4;
    const int l16  = lane & 15;
    const int wid  = blockIdx.x * 4 + (threadIdx.x >> 5);
    const int mt   = wid >> 6;  // 0..2047 : 16-row tile
    const int nt   = wid & 63;  // 0..63   : 64-col tile
    const int m0 = mt * 16;
    const int n0 = nt * 64;

    v8f acc[4];
#pragma unroll
    for (int q = 0; q < 4; ++q) {
        const int   col = n0 + 16 * q + l16;
        const float bv  = bih0[col] + bhh0[col];
#pragma unroll
        for (int j = 0; j < 8; ++j) acc[q][j] = bv;
    }

    const float* arow = X + (size_t)(m0 + l16) * Hsz;
    for (int k0 = 0; k0 < Hsz; k0 += 4) {
        v2f a = *(const v2f*)(arow + k0 + 2 * half);
#pragma unroll
        for (int q = 0; q < 4; ++q) {
            const float* brow = Wih0 + (size_t)(n0 + 16 * q + l16) * Hsz;
            v2f b = *(const v2f*)(brow + k0 + 2 * half);
            acc[q] = wmma_f32(a, b, acc[q]);
        }
    }

#pragma unroll
    for (int q = 0; q < 4; ++q)
#pragma unroll
        for (int j = 0; j < 8; ++j) {
            const int m = m0 + j + 8 * half;
            G0[(size_t)m * H4 + n0 + 16 * q + l16] = acc[q][j];
        }
}

// ===========================================================================
// Scan step, layer 0: gates = G0[:, t, :] + h0 @ W_hh0^T -> fused LSTM cell.
// Block: 128 threads = 4 waves = all 4 m-tiles of one nt. grid = 64 blocks.
// ===========================================================================
__global__ __launch_bounds__(128) void lstm_step_l0_kernel(
    const float* __restrict__ G0, const float* __restrict__ Whh0,
    const float* __restrict__ h_in, const float* __restrict__ c_in,
    float* __restrict__ h_out, float* __restrict__ c_out, int t) {
    __shared__ float sB[2 * BUFSZ];
    const int lane = threadIdx.x & 31;
    const int half = lane >> 4;
    const int l16  = lane & 15;
    const int wave = threadIdx.x >> 5;  // m-tile
    const int n0   = blockIdx.x * 16;   // shared n-tile
    const int m0   = wave * 16;

    v8f acc[4];
#pragma unroll
    for (int g = 0; g < 4; ++g)
#pragma unroll
        for (int j = 0; j < 8; ++j) {
            const int m = m0 + j + 8 * half;  // batch index
            acc[g][j]   = G0[((size_t)m * Ssz + t) * H4 + g * Hsz + n0 + l16];
        }

    accum_gates_lds(acc, Whh0, h_in, sB, n0);

#pragma unroll
    for (int j = 0; j < 8; ++j) {
        const int   m   = m0 + j + 8 * half;
        const int   col = n0 + l16;
        const float ig = sigf(acc[0][j]);
        const float fg = sigf(acc[1][j]);
        const float gg = tanhf(acc[2][j]);
        const float og = sigf(acc[3][j]);
        const float co = c_in[(size_t)m * Hsz + col];
        const float cn = fg * co + ig * gg;
        c_out[(size_t)m * Hsz + col] = cn;
        h_out[(size_t)m * Hsz + col] = og * tanhf(cn);
    }
}

// ===========================================================================
// Scan step, layer 1: gates = h0_new@W_ih1^T + h1@W_hh1^T + biases -> LSTM.
// ===========================================================================
__global__ __launch_bounds__(128) void lstm_step_l1_kernel(
    const float* __restrict__ Wih1, const float* __restrict__ Whh1,
    const float* __restrict__ bih1, const float* __restrict__ bhh1,
    const float* __restrict__ h0_new, const float* __restrict__ h_in,
    const float* __restrict__ c_in, float* __restrict__ h_out,
    float* __restrict__ c_out) {
    __shared__ float sB[2 * BUFSZ];
    const int lane = threadIdx.x & 31;
    const int half = lane >> 4;
    const int l16  = lane & 15;
    const int wave = threadIdx.x >> 5;
    const int n0   = blockIdx.x * 16;
    const int m0   = wave * 16;

    v8f acc[4];
#pragma unroll
    for (int g = 0; g < 4; ++g) {
        const int   col = g * Hsz + n0 + l16;
        const float bv  = bih1[col] + bhh1[col];
#pragma unroll
        for (int j = 0; j < 8; ++j) acc[g][j] = bv;
    }

    accum_gates_lds(acc, Wih1, h0_new, sB, n0);  // input-to-hidden (this step)
    accum_gates_lds(acc, Whh1, h_in, sB, n0);    // recurrent term

#pragma unroll
    for (int j = 0; j < 8; ++j) {
        const int   m   = m0 + j + 8 * half;
        const int   col = n0 + l16;
        const float ig = sigf(acc[0][j]);
        const float fg = sigf(acc[1][j]);
        const float gg = tanhf(acc[2][j]);
        const float og = sigf(acc[3][j]);
        const float co = c_in[(size_t)m * Hsz + col];
        const float cn = fg * co + ig * gg;
        c_out[(size_t)m * Hsz + col] = cn;
        h_out[(size_t)m * Hsz + col] = og * tanhf(cn);
    }
}

// ===========================================================================
// Head GEMM: out[64,1024] = [A0 | A1] @ W^T + bias  (optional ReLU)
// ===========================================================================
__global__ __launch_bounds__(128) void head_gemm_kernel(
    const float* __restrict__ A0, const float* __restrict__ A1,
    const float* __restrict__ W, const float* __restrict__ bias,
    float* __restrict__ out, int applyRelu) {
    const int lane = threadIdx.x & 31;
    const int half = lane >> 4;
    const int l16  = lane & 15;
    const int wid  = blockIdx.x * 4 + (threadIdx.x >> 5);
    const int mt   = wid >> 6;
    const int nt   = wid & 63;
    const int m0 = mt * 16;
    const int n0 = nt * 16;

    const int Ktot = (A1 != nullptr) ? 2 * Hsz : Hsz;

    v8f acc;
    {
        const float bv = bias[n0 + l16];
#pragma unroll
        for (int j = 0; j < 8; ++j) acc[j] = bv;
    }

    const float* wrow = W + (size_t)(n0 + l16) * Ktot;
    const float* ar0  = A0 + (size_t)(m0 + l16) * Hsz;
    for (int k0 = 0; k0 < Hsz; k0 += 4) {
        v2f a = *(const v2f*)(ar0 + k0 + 2 * half);
        v2f b = *(const v2f*)(wrow + k0 + 2 * half);
        acc   = wmma_f32(a, b, acc);
    }
    if (A1 != nullptr) {  // wave-uniform branch; EXEC stays all-ones
        const float* ar1 = A1 + (size_t)(m0 + l16) * Hsz;
        for (int k0 = 0; k0 < Hsz; k0 += 4) {
            v2f a = *(const v2f*)(ar1 + k0 + 2 * half);
            v2f b = *(const v2f*)(wrow + Hsz + k0 + 2 * half);
            acc   = wmma_f32(a, b, acc);
        }
    }

#pragma unroll
    for (int j = 0; j < 8; ++j) {
        const int m = m0 + j + 8 * half;
        float     v = acc[j];
        if (applyRelu) v = fmaxf(v, 0.0f);
        out[(size_t)m * Hsz + n0 + l16] = v;
    }
}

// ===========================================================================
// Row LayerNorm + sigmoid (+ optional elementwise multiplier)
// ===========================================================================
__global__ __launch_bounds__(256) void ln_sig_kernel(
    const float* __restrict__ y, const float* __restrict__ w,
    const float* __restrict__ bb, const float* __restrict__ mult,
    float* __restrict__ out) {
    __shared__ float s1[256];
    __shared__ float s2[256];
    const int m   = blockIdx.x;
    const int tid = threadIdx.x;
    float     a = 0.0f, b = 0.0f;
    for (int i = tid; i < Hsz; i += 256) {
        float v = y[(size_t)m * Hsz + i];
        a += v;
        b += v * v;
    }
    s1[tid] = a;
    s2[tid] = b;
    __syncthreads();
    for (int off = 128; off > 0; off >>= 1) {
        if (tid < off) {
            s1[tid] += s1[tid + off];
            s2[tid] += s2[tid + off];
        }
        __syncthreads();
    }
    const float mean = s1[0] * (1.0f / Hsz);
    const float var  = s2[0] * (1.0f / Hsz) - mean * mean;
    const float rstd = rsqrtf(var + 1e-5f);
    for (int i = tid; i < Hsz; i += 256) {
        float v = (y[(size_t)m * Hsz + i] - mean) * rstd * w[i] + bb[i];
        float o = sigf(v);
        if (mult != nullptr) o *= mult[(size_t)m * Hsz + i];
        out[(size_t)m * Hsz + i] = o;
    }
}

// state_importance_new = si + RATE * (mean_b |importance| - si)
__global__ __launch_bounds__(256) void si_update_kernel(
    const float* __restrict__ imp, const float* __restrict__ si,
    float* __restrict__ out) {
    const int n = blockIdx.x * 256 + threadIdx.x;
    if (n < Hsz) {
        float acc = 0.0f;
        for (int b = 0; b < Bsz; ++b) acc += fabsf(imp[(size_t)b * Hsz + n]);
        const float m = acc * (1.0f / Bsz);
        out[n]        = si[n] + 0.01f * (m - si[n]);
    }
}

__global__ __launch_bounds__(256) void zero_kernel(float* __restrict__ p, int n) {
    const int i = blockIdx.x * 256 + threadIdx.x;
    if (i < n) p[i] = 0.0f;
}

// ===========================================================================
extern "C" void kernel_launch(void* const* d_in, const int* in_sizes, int n_in,
                              void* d_out, int out_size, void* d_ws,
                              size_t ws_size, hipStream_t stream) {
    (void)in_sizes; (void)n_in; (void)out_size; (void)ws_size;

    const float* X    = (const float*)d_in[0];   // current_state [B,S,H]
    const float* ctx  = (const float*)d_in[1];   // context [B,H]
    const float* si   = (const float*)d_in[2];   // state_importance [1,H]
    //            d_in[3] memory_allocation (not returned)
    const float* W_ih = (const float*)d_in[4];   // [L,4H,H]
    const float* W_hh = (const float*)d_in[5];   // [L,4H,H]
    const float* b_ih = (const float*)d_in[6];   // [L,4H]
    const float* b_hh = (const float*)d_in[7];   // [L,4H]
    const float* Wg   = (const float*)d_in[8];   // [H,2H]
    const float* bg   = (const float*)d_in[9];   // [H]
    const float* lngw = (const float*)d_in[10];
    const float* lngb = (const float*)d_in[11];
    const float* W1   = (const float*)d_in[12];  // [H,H]
    const float* b1   = (const float*)d_in[13];
    const float* W2   = (const float*)d_in[14];  // [H,H]
    const float* b2   = (const float*)d_in[15];
    const float* lniw = (const float*)d_in[16];
    const float* lnib = (const float*)d_in[17];

    float* out = (float*)d_out;  // [B*H compressed][H state_importance_new]
    float* ws  = (float*)d_ws;

    // Workspace layout
    float* G0  = ws;                               // [B*S, 4H]
    size_t ofs = (size_t)Bsz * Ssz * H4;
    float* h0  = ws + ofs; ofs += 2 * BH;          // ping-pong states
    float* c0  = ws + ofs; ofs += 2 * BH;
    float* h1  = ws + ofs; ofs += 2 * BH;
    float* c1  = ws + ofs; ofs += 2 * BH;
    float* ygate  = ws + ofs; ofs += BH;
    float* hidden = ws + ofs; ofs += BH;
    float* zbuf   = ws + ofs; ofs += BH;
    float* impbuf = ws + ofs; ofs += BH;

    const float* Wih0 = W_ih;
    const float* Wih1 = W_ih + (size_t)H4 * Hsz;
    const float* Whh0 = W_hh;
    const float* Whh1 = W_hh + (size_t)H4 * Hsz;
    const float* bih1 = b_ih + H4;
    const float* bhh1 = b_hh + H4;

    // Zero initial LSTM states (h0,c0,h1,c1 contiguous: 8*BH floats)
    {
        const int n = 8 * BH;
        zero_kernel<<<(n + 255) / 256, 256, 0, stream>>>(h0, n);
    }

    // Phase 1: hoisted input-to-hidden GEMM for layer 0 (biases folded in)
    g0_gemm_kernel<<<(2048 * 64) / 4, 128, 0, stream>>>(X, Wih0, b_ih, b_hh, G0);

    // Phase 2: sequential scan; kernel boundaries provide grid-wide sync
    for (int t = 0; t < Ssz; ++t) {
        const int rd = t & 1;
        const int wr = (t + 1) & 1;
        lstm_step_l0_kernel<<<64, 128, 0, stream>>>(
            G0, Whh0, h0 + rd * BH, c0 + rd * BH, h0 + wr * BH, c0 + wr * BH, t);
        lstm_step_l1_kernel<<<64, 128, 0, stream>>>(
            Wih1, Whh1, bih1, bhh1, h0 + wr * BH, h1 + rd * BH, c1 + rd * BH,
            h1 + wr * BH, c1 + wr * BH);
    }
    const float* hfin = h1;  // final write slot = S & 1 == 0

    // Phase 3: head
    head_gemm_kernel<<<64, 128, 0, stream>>>(hfin, ctx, Wg, bg, ygate, 0);
    ln_sig_kernel<<<Bsz, 256, 0, stream>>>(ygate, lngw, lngb, hfin, out);
    head_gemm_kernel<<<64, 128, 0, stream>>>(out, nullptr, W1, b1, hidden, 1);
    head_gemm_kernel<<<64, 128, 0, stream>>>(hidden, nullptr, W2, b2, zbuf, 0);
    ln_sig_kernel<<<Bsz, 256, 0, stream>>>(zbuf, lniw, lnib, nullptr, impbuf);
    si_update_kernel<<<(Hsz + 255) / 256, 256, 0, stream>>>(impbuf, si, out + BH);
}